// PoneGNN_76544907149487
// MI455X (gfx1250) — compile-verified
//
#include <hip/hip_runtime.h>

#define NUM_U 100000
#define NUM_V 50000
#define NUM_NODES 150000
#define DIM 64
#define NUM_EDGES 2000000
#define NTILES (NUM_NODES / 16)   // 9375 node tiles of 16 rows x 64 dims
#define UTILES (NUM_U / 16)       // 6250 (tiles never straddle the user/item boundary)
#define ALPHA (1.0f / 3.0f)       // 1/(NUM_LAYER+1)

typedef __attribute__((ext_vector_type(2))) float v2f;
typedef __attribute__((ext_vector_type(8))) float v8f;

// acc_out = ALPHA * X + (Cin ? Cin : 0), done as 4 chained V_WMMA_F32_16X16X4_F32
// per 16x16 tile (A_g = X[:,4g:4g+4], B_g = alpha * identity slice) -> exact f32.
// Optionally copies the (split) seed source into seed0/seed1 (next-layer buffers).
// One wave handles one 16-node x 64-dim tile; X is staged via LDS for coalescing.
__global__ __launch_bounds__(256) void axpy_seed_wmma(
    const float* Xu, const float* Xi,          // acc source, split at NUM_U rows
    const float* Cin, float* accOut,           // may alias; Cin==nullptr -> zero C
    const float* __restrict__ Su, const float* __restrict__ Si,  // seed source (split)
    float* __restrict__ seed0, float* __restrict__ seed1)
{
    __shared__ __align__(16) float lds[8 * 1024];   // 4KB per wave, 8 waves/block
    const int lane = threadIdx.x & 31;
    const int tile = blockIdx.x * (blockDim.x >> 5) + (threadIdx.x >> 5);
    if (tile >= NTILES) return;   // wave-uniform: EXEC stays all-1s for WMMA

    float* myLds = lds + (threadIdx.x >> 5) * 1024;

    // ---- seed copies (fully coalesced float4 path) ----
    if (seed0) {
        const float4* ss = (const float4*)((tile < UTILES)
                               ? (Su + (size_t)tile * 1024)
                               : (Si + (size_t)(tile - UTILES) * 1024));
        float4* s0 = (float4*)(seed0 + (size_t)tile * 1024);
        float4* s1 = seed1 ? (float4*)(seed1 + (size_t)tile * 1024) : nullptr;
#pragma unroll
        for (int i = 0; i < 8; ++i) {
            float4 v = ss[lane + 32 * i];
            s0[lane + 32 * i] = v;
            if (s1) s1[lane + 32 * i] = v;
        }
    }

    // ---- stage X tile (16 rows x 64 dims = 1024 floats) into LDS ----
    const float4* xs = (const float4*)((tile < UTILES)
                           ? (Xu + (size_t)tile * 1024)
                           : (Xi + (size_t)(tile - UTILES) * 1024));
    float4* lx = (float4*)myLds;
#pragma unroll
    for (int i = 0; i < 8; ++i) lx[lane + 32 * i] = xs[lane + 32 * i];

    const int m    = lane & 15;          // A-matrix row (M)
    const int kk   = (lane >> 4) << 1;   // A/B K MSB group: 0 or 2
    const int half = lane >> 4;          // C/D row-half select
    const int n16  = lane & 15;          // C/D column (N)

    // B_g[k][n] = ALPHA * delta(n, 4g + k); B layout: VGPR j, lane l -> K=j+2*(l>>4), N=l&15
    v2f bfrag[4];
#pragma unroll
    for (int g = 0; g < 4; ++g) {
        bfrag[g].x = (n16 == (4 * g + kk + 0)) ? ALPHA : 0.0f;
        bfrag[g].y = (n16 == (4 * g + kk + 1)) ? ALPHA : 0.0f;
    }

#pragma unroll
    for (int c = 0; c < 4; ++c) {        // 4 column tiles of 16 dims
        const size_t base = ((size_t)tile * 16 + 8 * half) * DIM + c * 16 + n16;
        v8f acc;
        if (Cin) {
#pragma unroll
            for (int v = 0; v < 8; ++v) acc[v] = Cin[base + (size_t)v * DIM];
        } else {
#pragma unroll
            for (int v = 0; v < 8; ++v) acc[v] = 0.0f;
        }
        // D = sum_g A_g x B_g + C  ==  ALPHA * X_tile + C   (exact f32)
#pragma unroll
        for (int g = 0; g < 4; ++g) {
            // A 16x4 layout: lane l holds X[m][c*16 + 4g + kk + {0,1}]
            v2f a = *(const v2f*)(myLds + m * DIM + c * 16 + 4 * g + kk);
            acc = __builtin_amdgcn_wmma_f32_16x16x4_f32(
                false, a, false, bfrag[g], (short)0, acc, false, false);
        }
#pragma unroll
        for (int v = 0; v < 8; ++v) accOut[base + (size_t)v * DIM] = acc[v];
    }
}

// to[dst[e]] += from[src[e]] (rowwise, 64 f32). One wave per edge: 32 lanes x float2.
// `to` must be pre-seeded with x so the GIN self-term x[i] is included.
__global__ __launch_bounds__(256) void scatter_add(
    const int* __restrict__ edges,   // [2, NUM_EDGES]: row0=src, row1=dst
    const float* __restrict__ from, float* __restrict__ to)
{
    const unsigned gid = blockIdx.x * 256u + threadIdx.x;
    const unsigned e = gid >> 5;
    if (e >= NUM_EDGES) return;
    const int t = (gid & 31) << 1;
    const int s = edges[e];
    const int d = edges[NUM_EDGES + e];
    const float2 v = *(const float2*)(from + (size_t)s * DIM + t);
    float* dp = to + (size_t)d * DIM + t;
    atomicAdd(dp,     v.x);
    atomicAdd(dp + 1, v.y);
}

extern "C" void kernel_launch(void* const* d_in, const int* in_sizes, int n_in,
                              void* d_out, int out_size, void* d_ws, size_t ws_size,
                              hipStream_t stream) {
    const float* user_emb = (const float*)d_in[0];
    const float* item_emb = (const float*)d_in[1];
    const float* user_neg = (const float*)d_in[2];
    const float* item_neg = (const float*)d_in[3];
    const int*   pos_edges = (const int*)d_in[4];
    const int*   neg_edges = (const int*)d_in[5];

    float* acc_pos = (float*)d_out;
    float* acc_neg = acc_pos + (size_t)NUM_NODES * DIM;
    float* A = (float*)d_ws;                       // 38.4 MB each; ws needs 76.8 MB
    float* B = A + (size_t)NUM_NODES * DIM;

    const dim3 bW(256), gW((NTILES + 7) / 8);              // 8 waves/block, 1 tile/wave
    const dim3 bS(256), gS((NUM_EDGES * 32u + 255) / 256); // 1 wave/edge

    for (int br = 0; br < 2; ++br) {
        const int* edges = br ? neg_edges : pos_edges;
        float* acc = br ? acc_neg : acc_pos;
        const float* xu = br ? user_neg : user_emb;   // acc init source
        const float* xi = br ? item_neg : item_emb;

        // acc = alpha * ego_{pos|neg};  A = B = ego_pos (both branches propagate ego_pos)
        hipLaunchKernelGGL(axpy_seed_wmma, gW, bW, 0, stream,
                           xu, xi, (const float*)nullptr, acc,
                           user_emb, item_emb, A, B);
        // layer 1: B = e0 + agg(e0)
        hipLaunchKernelGGL(scatter_add, gS, bS, 0, stream, edges, A, B);
        // acc += alpha * e1 ;  A = e1
        hipLaunchKernelGGL(axpy_seed_wmma, gW, bW, 0, stream,
                           B, B + (size_t)NUM_U * DIM, acc, acc,
                           B, B + (size_t)NUM_U * DIM, A, (float*)nullptr);
        // layer 2: A = e1 + agg(e1)
        hipLaunchKernelGGL(scatter_add, gS, bS, 0, stream, edges, B, A);
        // acc += alpha * e2
        hipLaunchKernelGGL(axpy_seed_wmma, gW, bW, 0, stream,
                           A, A + (size_t)NUM_U * DIM, acc, acc,
                           (const float*)nullptr, (const float*)nullptr,
                           (float*)nullptr, (float*)nullptr);
    }
}